// Sampling_24352464569658
// MI455X (gfx1250) — compile-verified
//
#include <hip/hip_runtime.h>

// Problem constants (BATCH=128, LAT_DIM=512 from the reference).
#define LDIM 512
#define NBATCH 128
#define NS_ITERS 24
#define BC_FLOATS (2L * NBATCH * LDIM * LDIM)   // 67,108,864 floats (complex64 bc, interleaved)
#define Y_OFS BC_FLOATS

typedef __attribute__((ext_vector_type(2))) float v2f;
typedef __attribute__((ext_vector_type(4))) float v4f;
typedef __attribute__((ext_vector_type(8))) float v8f;
typedef __attribute__((ext_vector_type(4))) unsigned int v4u;
typedef __attribute__((ext_vector_type(4))) int v4i;
typedef __attribute__((ext_vector_type(8))) int v8i;

// ---------------------------------------------------------------------------
// Exact rounded-DFT entry: f(t) = round(cos(2*pi*t/512)) + i*round(sin(2*pi*t/512))
// Integer thresholds (no trig, exact): cos>=0.5 <=> t<=85 or t>=427;
// cos<=-0.5 <=> 171<=t<=341; sin>=0.5 <=> 43<=t<=213; sin<=-0.5 <=> 299<=t<=469.
// ---------------------------------------------------------------------------
__device__ __forceinline__ void dft_entry(int t, float& re, float& im) {
    re = (t <= 85 || t >= 427) ? 1.0f : ((t >= 171 && t <= 341) ? -1.0f : 0.0f);
    im = (t >= 43 && t <= 213) ? 1.0f : ((t >= 299 && t <= 469) ? -1.0f : 0.0f);
}

// ---------------------------------------------------------------------------
// 1) HBM-bound zero fill of the bc region (268 MB). Nontemporal 128-bit stores.
// ---------------------------------------------------------------------------
__global__ void __launch_bounds__(256) fill_zero(v4f* __restrict__ p, long n4) {
    long stride = (long)gridDim.x * blockDim.x;
    v4f z = {0.0f, 0.0f, 0.0f, 0.0f};
    for (long i = (long)blockIdx.x * blockDim.x + threadIdx.x; i < n4; i += stride)
        __builtin_nontemporal_store(z, p + i);
}

// ---------------------------------------------------------------------------
// 2) Build M = rounded DFT (planar re/im, 512x512 f32 each; symmetric).
// ---------------------------------------------------------------------------
__global__ void __launch_bounds__(256) build_m(float* __restrict__ Mre, float* __restrict__ Mim) {
    int idx = blockIdx.x * blockDim.x + threadIdx.x;   // 262144 threads
    int j = idx >> 9, k = idx & 511;
    float re, im;
    dft_entry((j * k) & 511, re, im);
    Mre[idx] = re;
    Mim[idx] = im;
}

// ---------------------------------------------------------------------------
// 3) Row absolute sums (M symmetric => ||M||_1 == ||M||_inf).
// ---------------------------------------------------------------------------
__global__ void __launch_bounds__(256) row_abs_sum(const float* __restrict__ Mre,
                                                   const float* __restrict__ Mim,
                                                   float* __restrict__ rowsum) {
    __shared__ float s[256];
    int j = blockIdx.x;
    float acc = 0.0f;
    for (int k = threadIdx.x; k < LDIM; k += 256) {
        float re = Mre[j * LDIM + k], im = Mim[j * LDIM + k];
        acc += sqrtf(re * re + im * im);
    }
    s[threadIdx.x] = acc;
    __syncthreads();
    for (int off = 128; off > 0; off >>= 1) {
        if (threadIdx.x < off) s[threadIdx.x] += s[threadIdx.x + off];
        __syncthreads();
    }
    if (threadIdx.x == 0) rowsum[j] = s[0];
}

__global__ void __launch_bounds__(512) max_scale(const float* __restrict__ rowsum,
                                                 float* __restrict__ scale) {
    __shared__ float s[512];
    s[threadIdx.x] = rowsum[threadIdx.x];
    __syncthreads();
    for (int off = 256; off > 0; off >>= 1) {
        if (threadIdx.x < off) s[threadIdx.x] = fmaxf(s[threadIdx.x], s[threadIdx.x + off]);
        __syncthreads();
    }
    if (threadIdx.x == 0) scale[0] = 1.0f / (s[0] * s[0]);
}

// ---------------------------------------------------------------------------
// 4) X0 = M^H / (||M||_1 ||M||_inf) = conj(M) * scale   (M symmetric)
// ---------------------------------------------------------------------------
__global__ void __launch_bounds__(256) init_x(const float* __restrict__ Mre,
                                              const float* __restrict__ Mim,
                                              const float* __restrict__ scale,
                                              float* __restrict__ Xre, float* __restrict__ Xim) {
    int idx = blockIdx.x * blockDim.x + threadIdx.x;
    float sc = scale[0];
    Xre[idx] = Mre[idx] * sc;
    Xim[idx] = -Mim[idx] * sc;
}

// ---------------------------------------------------------------------------
// Build a 2-D TDM descriptor group-0 for a 16x512 f32 tile -> LDS.
// Layout per CDNA5 ISA: w0 = count(=1), w1 = lds_addr,
// w2 = global_addr[31:0], w3 = global_addr[56:32] | type(2)<<30.
// ---------------------------------------------------------------------------
__device__ __forceinline__ v4u tdm_group0(unsigned lds_addr, const float* gptr) {
    unsigned long ga = (unsigned long)gptr;
    v4u g0;
    g0[0] = 1u;                                   // count=1, user descriptor
    g0[1] = lds_addr;                             // LDS byte address
    g0[2] = (unsigned)(ga & 0xffffffffu);
    g0[3] = (unsigned)((ga >> 32) & 0x01ffffffu) | (2u << 30);  // type=2 ("image")
    return g0;
}

// ---------------------------------------------------------------------------
// 5) Complex 512x512 GEMM, C = A*B, via v_wmma_f32_16x16x4_f32.
//    - All 8 waves of a block share one 16-row A band: wave 0 stages it
//      (re+im planes, 2x32 KB) into LDS with the Tensor Data Mover
//      (tensor_load_to_lds + s_wait_tensorcnt), then every wave reads its
//      A fragments with ds_load_b64.  B fragments come from L2 (global).
//    - The TDM descriptor carries the LDS address as an integer, so the
//      compiler cannot see the write to lds_a; an asm escape with a memory
//      clobber keeps the LDS reads from being folded to undef.
//    - Per K-step of 4: 4 WMMA ops. F32 WMMA forbids A/B NEG modifiers, so
//      the -Aim*Bim term uses a VALU-negated A fragment.
//    A 16x4 f32 operand: lane(0..15)=row, reg pair holds K={2h,2h+1}, h=lane>>4.
//    B 4x16 operand mirrors K split; C/D: reg v -> row (tm + v + 8h), col lane&15.
// ---------------------------------------------------------------------------
__global__ void __launch_bounds__(256) cgemm512(const float* __restrict__ Are,
                                                const float* __restrict__ Aim,
                                                const float* __restrict__ Bre,
                                                const float* __restrict__ Bim,
                                                float* __restrict__ Cre,
                                                float* __restrict__ Cim) {
    __shared__ float lds_a[2 * 16 * LDIM];        // 64 KB: [0..8191]=re band, [8192..]=im band

    const int lane = threadIdx.x & 31;
    const int wave = threadIdx.x >> 5;
    const int tm = (blockIdx.x >> 2) << 4;        // A row band, shared by all 8 waves
    const int tn = ((blockIdx.x * 8 + wave) & 31) << 4;   // C col base per wave
    const int half = lane >> 4;
    const int l15 = lane & 15;
    const int bcol = tn + l15;

    // --- TDM stage of the A band (16 rows x 512 cols, re and im planes) ---
    if (wave == 0) {
        v8i g1;
        g1[0] = (int)(2u << 16);          // data_size = 4 bytes
        g1[1] = (int)(512u << 16);        // tensor_dim0 = 512 (row length)
        g1[2] = (int)(512u << 16);        // tensor_dim1 = 512 (rows)
        g1[3] = (int)(512u << 16);        // tile_dim0 = 512
        g1[4] = 16;                       // tile_dim1 = 16 rows
        g1[5] = 512;                      // tensor_dim0_stride = 512 elements
        g1[6] = 0;
        g1[7] = 0;
        v4i gz4 = {0, 0, 0, 0};
        v8i gz8 = {0, 0, 0, 0, 0, 0, 0, 0};
        __builtin_amdgcn_tensor_load_to_lds(tdm_group0(0u, Are + tm * LDIM),
                                            g1, gz4, gz4, gz8, 0);
        __builtin_amdgcn_tensor_load_to_lds(tdm_group0(32768u, Aim + tm * LDIM),
                                            g1, gz4, gz4, gz8, 0);
        __builtin_amdgcn_s_wait_tensorcnt(0);
    }
    __syncthreads();
    // The TDM wrote lds_a behind the compiler's back: escape the pointer with
    // a memory clobber so the subsequent ds_load reads are not treated as
    // loads of uninitialized (undef) memory.
    asm volatile("" : : "r"(&lds_a[0]) : "memory");

    const float* lre = lds_a + l15 * LDIM;            // A row l15, re
    const float* lim = lds_a + 16 * LDIM + l15 * LDIM;  // A row l15, im

    v8f acc_re = {};
    v8f acc_im = {};

    for (int k = 0; k < LDIM; k += 4) {
        if ((k & 63) == 0 && k + 64 < LDIM)           // -> global_prefetch_b8
            __builtin_prefetch(Bre + (k + 64) * LDIM + bcol, 0, 1);
        const int ka = k + 2 * half;
        v2f a_re = *(const v2f*)(lre + ka);           // ds_load_b64 (8B aligned)
        v2f a_im = *(const v2f*)(lim + ka);
        v2f a_im_neg = -a_im;                         // VALU negate (F32 WMMA has no A-neg)
        v2f b_re, b_im;
        b_re.x = Bre[ka * LDIM + bcol];
        b_re.y = Bre[(ka + 1) * LDIM + bcol];
        b_im.x = Bim[ka * LDIM + bcol];
        b_im.y = Bim[(ka + 1) * LDIM + bcol];

        // acc_re += Are*Bre - Aim*Bim ; acc_im += Are*Bim + Aim*Bre
        acc_re = __builtin_amdgcn_wmma_f32_16x16x4_f32(false, a_re, false, b_re,
                                                       (short)0, acc_re, false, false);
        acc_re = __builtin_amdgcn_wmma_f32_16x16x4_f32(false, a_im_neg, false, b_im,
                                                       (short)0, acc_re, false, false);
        acc_im = __builtin_amdgcn_wmma_f32_16x16x4_f32(false, a_re, false, b_im,
                                                       (short)0, acc_im, false, false);
        acc_im = __builtin_amdgcn_wmma_f32_16x16x4_f32(false, a_im, false, b_re,
                                                       (short)0, acc_im, false, false);
    }

#pragma unroll
    for (int v = 0; v < 8; ++v) {
        int row = tm + v + 8 * half;
        Cre[row * LDIM + bcol] = acc_re[v];
        Cim[row * LDIM + bcol] = acc_im[v];
    }
}

// ---------------------------------------------------------------------------
// 6) T <- 2I - T (complex, elementwise + diagonal shift)
// ---------------------------------------------------------------------------
__global__ void __launch_bounds__(256) fix_t(float* __restrict__ Tre, float* __restrict__ Tim) {
    int idx = blockIdx.x * blockDim.x + threadIdx.x;
    int j = idx >> 9, k = idx & 511;
    float d = (j == k) ? 2.0f : 0.0f;
    Tre[idx] = d - Tre[idx];
    Tim[idx] = -Tim[idx];
}

// ---------------------------------------------------------------------------
// 7) Scatter the 639 nonzeros of bc:
//    bc[r,0]      = dd[0]*eig[r,0]      r<127   (dd[0] = X[0,0], dft_diag[0]=1)
//    bc[127+l, l] = dd[l]*eig[127,l]            (dd[l] = dft_diag[l]*X[l,l])
// ---------------------------------------------------------------------------
__global__ void __launch_bounds__(512) finalize_bc(const float* __restrict__ Xre,
                                                   const float* __restrict__ Xim,
                                                   const float* __restrict__ eig,
                                                   float* __restrict__ out) {
    int l = threadIdx.x;   // 0..511
    float dr, di;
    dft_entry((l * l) & 511, dr, di);
    float xr = Xre[l * LDIM + l], xi = Xim[l * LDIM + l];
    float ddr = dr * xr - di * xi;
    float ddi = dr * xi + di * xr;
    float e = eig[(NBATCH - 1) * LDIM + l];
    long c = (long)(NBATCH - 1 + l) * LDIM + l;      // complex index
    out[2 * c]     = ddr * e;
    out[2 * c + 1] = ddi * e;
    if (l < NBATCH - 1) {
        float x0r = Xre[0], x0i = Xim[0];
        float e0 = eig[l * LDIM];
        long c0 = (long)l * LDIM;
        out[2 * c0]     = x0r * e0;
        out[2 * c0 + 1] = x0i * e0;
    }
}

// ---------------------------------------------------------------------------
// 8) Y[b,l] = Re(q[l])*a_eps[l,b] - Im(q[l])*b_eps[l,b], q = dft_diag*sqrt(eig[127])
// ---------------------------------------------------------------------------
__global__ void __launch_bounds__(256) compute_y(const float* __restrict__ eig,
                                                 const float* __restrict__ a_eps,
                                                 const float* __restrict__ b_eps,
                                                 float* __restrict__ out) {
    int idx = blockIdx.x * blockDim.x + threadIdx.x;  // 65536
    int b = idx >> 9, l = idx & 511;
    float dr, di;
    dft_entry((l * l) & 511, dr, di);
    float s = sqrtf(eig[(NBATCH - 1) * LDIM + l]);
    float qr = dr * s, qi = di * s;
    float av = a_eps[l * NBATCH + b];
    float bv = b_eps[l * NBATCH + b];
    out[Y_OFS + (long)b * LDIM + l] = qr * av - qi * bv;
}

// ---------------------------------------------------------------------------
extern "C" void kernel_launch(void* const* d_in, const int* in_sizes, int n_in,
                              void* d_out, int out_size, void* d_ws, size_t ws_size,
                              hipStream_t stream) {
    (void)in_sizes; (void)n_in; (void)out_size; (void)ws_size;
    const float* eig   = (const float*)d_in[0];   // (128,512)
    const float* a_eps = (const float*)d_in[1];   // (512,128)
    const float* b_eps = (const float*)d_in[2];   // (512,128)
    float* out = (float*)d_out;
    float* ws  = (float*)d_ws;

    const long NN = (long)LDIM * LDIM;            // 262144
    float* Mre = ws;            float* Mim = ws + NN;
    float* Xre = ws + 2 * NN;   float* Xim = ws + 3 * NN;
    float* Tre = ws + 4 * NN;   float* Tim = ws + 5 * NN;
    float* Ure = ws + 6 * NN;   float* Uim = ws + 7 * NN;
    float* rowsum = ws + 8 * NN;
    float* scale  = rowsum + LDIM;

    // HBM-bound part: zero the 268 MB bc region (16,777,216 x 16B NT stores).
    fill_zero<<<8192, 256, 0, stream>>>((v4f*)out, BC_FLOATS / 4);

    // Build rounded DFT and Newton-Schulz starting point (all L2-resident).
    build_m<<<NN / 256, 256, 0, stream>>>(Mre, Mim);
    row_abs_sum<<<LDIM, 256, 0, stream>>>(Mre, Mim, rowsum);
    max_scale<<<1, 512, 0, stream>>>(rowsum, scale);
    init_x<<<NN / 256, 256, 0, stream>>>(Mre, Mim, scale, Xre, Xim);

    // Newton-Schulz: X <- X(2I - M X). 24 fixed iterations (deterministic).
    for (int it = 0; it < NS_ITERS; ++it) {
        cgemm512<<<128, 256, 0, stream>>>(Mre, Mim, Xre, Xim, Tre, Tim);  // T = M*X
        fix_t<<<NN / 256, 256, 0, stream>>>(Tre, Tim);                    // T = 2I - T
        cgemm512<<<128, 256, 0, stream>>>(Xre, Xim, Tre, Tim, Ure, Uim);  // U = X*T
        float* t;
        t = Xre; Xre = Ure; Ure = t;
        t = Xim; Xim = Uim; Uim = t;
    }

    // Scatter nonzeros and compute Y.
    finalize_bc<<<1, 512, 0, stream>>>(Xre, Xim, eig, out);
    compute_y<<<256, 256, 0, stream>>>(eig, a_eps, b_eps, out);
}